// HetersparseGAT_74732430950521
// MI455X (gfx1250) — compile-verified
//
#include <hip/hip_runtime.h>
#include <hip/hip_bf16.h>

typedef float v2f __attribute__((ext_vector_type(2)));
typedef float v8f __attribute__((ext_vector_type(8)));

// ---------------------------------------------------------------------------
// Generic 64-K fp32 WMMA GEMM:  C[m, n] = sum_k A[m,k] * W(k,n),  Ncols = 64
//   WKMAJOR=1 : W(k,n) = W[k*64 + n]      (for hp = h @ w, w is [K][O])
//   WKMAJOR=0 : W(k,n) = W[n*64 + k]      (for X @ Wt.T, W is [O][K])
// Compile-time WKMAJOR -> branch-free inner loop (loads + v_wmma only).
// One wave computes one 16x16 tile via 16x V_WMMA_F32_16X16X4_F32 (full fp32).
// ---------------------------------------------------------------------------
template <int WKMAJOR>
__global__ __launch_bounds__(128) void gemm64_wmma_f32(
    const float* __restrict__ A, int lda, const float* __restrict__ W,
    float* __restrict__ C, int ldc, int M) {
  const int wave = threadIdx.x >> 5;   // 0..3 -> column tile (4*16 = 64 cols)
  const int lane = threadIdx.x & 31;
  const int half = lane >> 4;          // 0: K pair {0,1}; 1: K pair {2,3}
  const int l16  = lane & 15;
  const int rowBase = blockIdx.x * 16;
  const int colBase = wave * 16;

  const int row     = rowBase + l16;
  const float rMask = (row < M) ? 1.0f : 0.0f;
  const float* Arow = A + (size_t)(row < M ? row : 0) * lda;

  v8f acc = {};
#pragma unroll
  for (int kk = 0; kk < 64; kk += 4) {
    const int k0 = kk + 2 * half;
    v2f a, b;
    // A fragment: contiguous pair -> global_load_b64; value-mask (no EXEC div)
    a = *(const v2f*)(Arow + k0);
    a.x *= rMask;
    a.y *= rMask;
    if (WKMAJOR) {
      // W[k][n]: stride-64 between the two k values -> two b32 loads
      b.x = W[(size_t)(k0)     * 64 + colBase + l16];
      b.y = W[(size_t)(k0 + 1) * 64 + colBase + l16];
    } else {
      // W[n][k]: the two k values are contiguous -> one b64 load
      b = *(const v2f*)(W + (size_t)(colBase + l16) * 64 + k0);
    }
    acc = __builtin_amdgcn_wmma_f32_16x16x4_f32(false, a, false, b, (short)0,
                                                acc, false, false);
  }
#pragma unroll
  for (int g = 0; g < 8; ++g) {
    const int r = rowBase + g + 8 * half;   // C/D layout: vgpr g -> M=g / M=g+8
    if (r < M) C[(size_t)r * ldc + colBase + l16] = acc[g];
  }
}

// ---------------------------------------------------------------------------
// attn_src/attn_trg per node: dot(hp[n][h][:], a[h][:]), wave per node.
// ---------------------------------------------------------------------------
__global__ __launch_bounds__(256) void attn_kernel(
    const float* __restrict__ hp, const float* __restrict__ a_src,
    const float* __restrict__ a_trg, float* __restrict__ as_out,
    float* __restrict__ at_out, int n) {
  const int lane = threadIdx.x & 31;
  const int node = blockIdx.x * 8 + (threadIdx.x >> 5);
  if (node >= n) return;
  const float* base = hp + (size_t)node * 128;
#pragma unroll
  for (int h = 0; h < 2; ++h) {
    float s = 0.f, t = 0.f;
    for (int f = lane; f < 64; f += 32) {
      const float v = base[h * 64 + f];
      s += v * a_src[h * 64 + f];
      t += v * a_trg[h * 64 + f];
    }
    for (int off = 16; off; off >>= 1) {
      s += __shfl_xor(s, off, 32);
      t += __shfl_xor(t, off, 32);
    }
    if (lane == 0) { as_out[node * 2 + h] = s; at_out[node * 2 + h] = t; }
  }
}

__device__ __forceinline__ unsigned orderF(float f) {
  unsigned u = __float_as_uint(f);
  return (u & 0x80000000u) ? ~u : (u | 0x80000000u);
}
__device__ __forceinline__ float deorderF(unsigned o) {
  return __uint_as_float((o & 0x80000000u) ? (o & 0x7FFFFFFFu) : ~o);
}

// Global max of leaky_relu(as[src]+at[trg]) over all (edge, head).
__global__ __launch_bounds__(256) void edge_max_kernel(
    const float* __restrict__ as_, const float* __restrict__ at_,
    const int* __restrict__ src, const int* __restrict__ trg,
    unsigned* __restrict__ gmax, int nE) {
  const int e = blockIdx.x * 256 + threadIdx.x;
  float m = -1e30f;
  if (e < nE) {
    const int s = src[e], t = trg[e];
#pragma unroll
    for (int h = 0; h < 2; ++h) {
      float v = as_[(size_t)s * 2 + h] + at_[(size_t)t * 2 + h];
      v = v > 0.f ? v : 0.2f * v;
      m = fmaxf(m, v);
    }
  }
  __shared__ float red[256];
  red[threadIdx.x] = m;
  __syncthreads();
  for (int stride = 128; stride; stride >>= 1) {
    if (threadIdx.x < stride)
      red[threadIdx.x] = fmaxf(red[threadIdx.x], red[threadIdx.x + stride]);
    __syncthreads();
  }
  if (threadIdx.x == 0) atomicMax(gmax, orderF(red[0]));
}

// exp(e - max) per (edge,head) + segment-sum denominators into denom[trg][h].
__global__ __launch_bounds__(256) void edge_exp_kernel(
    const float* __restrict__ as_, const float* __restrict__ at_,
    const int* __restrict__ src, const int* __restrict__ trg,
    const unsigned* __restrict__ gmax, float* __restrict__ expE,
    float* __restrict__ denom, int nE) {
  const int e = blockIdx.x * 256 + threadIdx.x;
  if (e >= nE) return;
  const float M = deorderF(*gmax);
  const int s = src[e], t = trg[e];
#pragma unroll
  for (int h = 0; h < 2; ++h) {
    float v = as_[(size_t)s * 2 + h] + at_[(size_t)t * 2 + h];
    v = v > 0.f ? v : 0.2f * v;
    const float ex = expf(v - M);
    expE[(size_t)e * 2 + h] = ex;
    unsafeAtomicAdd(&denom[(size_t)t * 2 + h], ex);
  }
}

// Heavy phase: out[trg][h][f] += hp[src][h][f] * attn.  64 lanes per
// (edge,head) -> coalesced cacheline gathers + f32 atomic adds (L2-resident).
__global__ __launch_bounds__(256) void edge_scatter_kernel(
    const float* __restrict__ hp, const float* __restrict__ expE,
    const float* __restrict__ denom, const int* __restrict__ src,
    const int* __restrict__ trg, float* __restrict__ out, int nE) {
  const long long pair = (long long)blockIdx.x * 4 + (threadIdx.x >> 6);
  const int f = threadIdx.x & 63;
  const long long e = pair >> 1;
  const int h = (int)(pair & 1);
  if (e >= nE) return;
  const int s = src[e], t = trg[e];
  const float attn = expE[e * 2 + h] / (denom[(size_t)t * 2 + h] + 1e-16f);
  const float v = hp[(size_t)s * 128 + h * 64 + f] * attn;
  unsafeAtomicAdd(&out[(size_t)t * 128 + h * 64 + f], v);
}

// x[n][f] = mean over heads (first n_user nodes only are consumed downstream)
__global__ __launch_bounds__(256) void mean_heads_kernel(
    const float* __restrict__ out, float* __restrict__ x, int n) {
  const int i = blockIdx.x * 256 + threadIdx.x;
  if (i >= n * 64) return;
  const int node = i >> 6, f = i & 63;
  x[i] = 0.5f * (out[(size_t)node * 128 + f] + out[(size_t)node * 128 + 64 + f]);
}

// Fusion epilogue, wave per node: tanh-attention over the 2 types, softmax
// betas, fused concat @ fc_w, log_softmax.
__global__ __launch_bounds__(256) void fusion_kernel(
    const float* __restrict__ Abuf, const float* __restrict__ B0,
    const float* __restrict__ B1, const float* __restrict__ x0,
    const float* __restrict__ x1, const float* __restrict__ aa_m,
    const float* __restrict__ fc_w, const float* __restrict__ fc_b,
    float* __restrict__ outp, int nU) {
  const int lane = threadIdx.x & 31;
  const int node = blockIdx.x * 8 + (threadIdx.x >> 5);
  if (node >= nU) return;
  const size_t b = (size_t)node * 64;
  float s0 = 0.f, s1 = 0.f;
  for (int f = lane; f < 64; f += 32) {
    const float a = Abuf[b + f], m = aa_m[f];
    s0 += tanhf(a + B0[b + f]) * m;
    s1 += tanhf(a + B1[b + f]) * m;
  }
  for (int off = 16; off; off >>= 1) {
    s0 += __shfl_xor(s0, off, 32);
    s1 += __shfl_xor(s1, off, 32);
  }
  const float mx = fmaxf(s0, s1);
  const float e0 = expf(s0 - mx), e1 = expf(s1 - mx);
  const float inv = 1.0f / (e0 + e1);
  const float b0 = e0 * inv, b1 = e1 * inv;
  float l0 = 0.f, l1 = 0.f;
  for (int f = lane; f < 64; f += 32) {
    const float t0 = x0[b + f], t1 = x1[b + f];
    const float fu = b0 * t0 + b1 * t1;
    l0 += t0 * fc_w[f]       + t1 * fc_w[64 + f]  + fu * fc_w[128 + f];
    l1 += t0 * fc_w[192 + f] + t1 * fc_w[256 + f] + fu * fc_w[320 + f];
  }
  for (int off = 16; off; off >>= 1) {
    l0 += __shfl_xor(l0, off, 32);
    l1 += __shfl_xor(l1, off, 32);
  }
  if (lane == 0) {
    l0 += fc_b[0]; l1 += fc_b[1];
    const float m2 = fmaxf(l0, l1);
    const float lse = m2 + logf(expf(l0 - m2) + expf(l1 - m2));
    outp[(size_t)node * 2 + 0] = l0 - lse;
    outp[(size_t)node * 2 + 1] = l1 - lse;
  }
}

extern "C" void kernel_launch(void* const* d_in, const int* in_sizes, int n_in,
                              void* d_out, int out_size, void* d_ws,
                              size_t ws_size, hipStream_t stream) {
  const float* h     = (const float*)d_in[0];
  const int*   srcE[2] = {(const int*)d_in[1], (const int*)d_in[3]};
  const int*   trgE[2] = {(const int*)d_in[2], (const int*)d_in[4]};
  const float* wL[2]    = {(const float*)d_in[5], (const float*)d_in[8]};
  const float* asrcL[2] = {(const float*)d_in[6], (const float*)d_in[9]};
  const float* atrgL[2] = {(const float*)d_in[7], (const float*)d_in[10]};
  const float* aa_w1 = (const float*)d_in[11];
  const float* aa_w2 = (const float*)d_in[12];
  const float* aa_m  = (const float*)d_in[13];
  const float* fc_w  = (const float*)d_in[14];
  const float* fc_b  = (const float*)d_in[15];

  const int N  = in_sizes[0] / 64;          // 100000
  const int EL[2] = {in_sizes[1], in_sizes[3]};
  const int NU = (out_size > 0) ? out_size / 2 : 90000;  // n_user (output rows)

  // ---- workspace bump layout (256B aligned) ----
  char* ws = (char*)d_ws;
  size_t off = 0;
  auto alloc = [&](size_t bytes) -> size_t {
    size_t o = off;
    off += (bytes + 255) & ~(size_t)255;
    return o;
  };
  const size_t szHP = (size_t)N * 128 * sizeof(float);
  const size_t o_hp  = alloc(szHP);
  const size_t o_out = alloc(szHP);
  const size_t o_as  = alloc((size_t)N * 2 * sizeof(float));
  const size_t o_at  = alloc((size_t)N * 2 * sizeof(float));
  const size_t o_dn  = alloc((size_t)N * 2 * sizeof(float));
  const size_t maxE  = (size_t)(EL[0] > EL[1] ? EL[0] : EL[1]);
  const size_t o_exp = alloc(maxE * 2 * sizeof(float));
  const size_t o_gm  = alloc(sizeof(unsigned));
  const size_t o_x0  = alloc((size_t)NU * 64 * sizeof(float));
  const size_t o_x1  = alloc((size_t)NU * 64 * sizeof(float));

  float* hp   = (float*)(ws + o_hp);
  float* oacc = (float*)(ws + o_out);
  float* as_  = (float*)(ws + o_as);
  float* at_  = (float*)(ws + o_at);
  float* dn   = (float*)(ws + o_dn);
  float* expE = (float*)(ws + o_exp);
  unsigned* gm = (unsigned*)(ws + o_gm);
  float* xL[2] = {(float*)(ws + o_x0), (float*)(ws + o_x1)};
  // phase-3 GEMM outputs alias the dead hp/out region (2*szHP >= 3*NU*64*4)
  const size_t szX = ((size_t)NU * 64 * sizeof(float) + 255) & ~(size_t)255;
  float* Abuf = (float*)(ws + o_hp);
  float* B0   = (float*)(ws + o_hp + szX);
  float* B1   = (float*)(ws + o_hp + 2 * szX);

  const int mTilesN  = (N + 15) / 16;
  const int mTilesNU = (NU + 15) / 16;

  for (int l = 0; l < 2; ++l) {
    const int E = EL[l];
    hipMemsetAsync(oacc, 0, szHP, stream);
    hipMemsetAsync(dn, 0, (size_t)N * 2 * sizeof(float), stream);
    hipMemsetAsync(gm, 0, sizeof(unsigned), stream);
    for (int head = 0; head < 2; ++head)  // hp[:,head,:] = h @ w[head]
      gemm64_wmma_f32<1><<<mTilesN, 128, 0, stream>>>(
          h, 64, wL[l] + (size_t)head * 4096, hp + head * 64, 128, N);
    attn_kernel<<<(N + 7) / 8, 256, 0, stream>>>(hp, asrcL[l], atrgL[l], as_,
                                                 at_, N);
    edge_max_kernel<<<(E + 255) / 256, 256, 0, stream>>>(as_, at_, srcE[l],
                                                         trgE[l], gm, E);
    edge_exp_kernel<<<(E + 255) / 256, 256, 0, stream>>>(
        as_, at_, srcE[l], trgE[l], gm, expE, dn, E);
    edge_scatter_kernel<<<(2 * E + 3) / 4, 256, 0, stream>>>(
        hp, expE, dn, srcE[l], trgE[l], oacc, E);
    mean_heads_kernel<<<(NU * 64 + 255) / 256, 256, 0, stream>>>(oacc, xL[l],
                                                                 NU);
  }

  // Phase 3: three 64x64 fp32 WMMA GEMMs + fused epilogue
  gemm64_wmma_f32<0><<<mTilesNU, 128, 0, stream>>>(h, 64, aa_w1, Abuf, 64, NU);
  gemm64_wmma_f32<0><<<mTilesNU, 128, 0, stream>>>(xL[0], 64, aa_w2, B0, 64, NU);
  gemm64_wmma_f32<0><<<mTilesNU, 128, 0, stream>>>(xL[1], 64, aa_w2, B1, 64, NU);
  fusion_kernel<<<(NU + 7) / 8, 256, 0, stream>>>(
      Abuf, B0, B1, xL[0], xL[1], aa_m, fc_w, fc_b, (float*)d_out, NU);
}